// DeformConv3d_90237262889599
// MI455X (gfx1250) — compile-verified
//
#include <hip/hip_runtime.h>
#include <hip/hip_bf16.h>
#include <cstdint>

typedef float v2f __attribute__((ext_vector_type(2)));
typedef float v4f __attribute__((ext_vector_type(4)));
typedef float v8f __attribute__((ext_vector_type(8)));

#define BB   4
#define CIN  64
#define COUT 64
#define DDIM 8
#define HDIM 48
#define WDIM 48
#define NN   (DDIM*HDIM*WDIM)   /* 18432 */
#define KK   27
#define KCH  108                /* 81 offset + 27 mask conv channels */
#define HALO_ROWS 162           /* 3z * 3y * 18x */
#define SXS  68                 /* padded cin stride (bank-conflict-free b64 frags) */
#define OFFP 17                 /* padded pos stride for conv-result tile */

/* dynamic LDS layout (floats):
   s_x   : 162*68 = 11016   (halo of xT, [row][cin])
   s_off : 112*17 =  1904   (conv result, [ch][pos])
   s_cw  : 432*8  =  3456   (corner weights * valid * mask)
   s_ci  : 432*8  =  3456   (corner spatial indices, int)
   s_a   : 16*68  =  1088   (sampled A tile, [pos][cin])            */
#define SMEM_FLOATS (11016 + 1904 + 3456 + 3456 + 1088)

// ---- CDNA5 async global->LDS b128 copy (ASYNCcnt path), builtin or asm ----
__device__ __forceinline__ void async_copy_b128(const float* gsrc, float* ldst) {
#if __has_builtin(__builtin_amdgcn_global_load_async_to_lds_b128)
    typedef int a4i __attribute__((vector_size(16)));          // matches builtin's pointee
    typedef __attribute__((address_space(1))) a4i* gp_t;       // global (device) ptr
    typedef __attribute__((address_space(3))) a4i* lp_t;       // LDS (shared) ptr, 32-bit
    __builtin_amdgcn_global_load_async_to_lds_b128(
        (gp_t)(uintptr_t)gsrc,
        (lp_t)(uint32_t)(uintptr_t)ldst,   // low 32 bits of generic = LDS byte addr
        0, 0);
#else
    const uint32_t loff = (uint32_t)(uintptr_t)ldst;
    asm volatile("global_load_async_to_lds_b128 %0, %1, off"
                 :: "v"(loff), "v"(gsrc) : "memory");
#endif
}

__device__ __forceinline__ void wait_async_zero() {
#if __has_builtin(__builtin_amdgcn_s_wait_asynccnt)
    __builtin_amdgcn_s_wait_asynccnt(0);
#else
    asm volatile("s_wait_asynccnt 0" ::: "memory");
#endif
}

__global__ __launch_bounds__(256)
void transpose_x_kernel(const float* __restrict__ x, float* __restrict__ xT) {
    __shared__ float s[16][17];
    const int bx = blockIdx.x;
    const int ct = bx & 3;                 // cin tile (64/16)
    const int nt = (bx >> 2) % (NN / 16);  // n tile
    const int b  = bx / (4 * (NN / 16));
    const int tx = threadIdx.x & 15;
    const int ty = threadIdx.x >> 4;
    const int n0 = nt * 16, c0 = ct * 16;
    s[ty][tx] = x[((size_t)b * CIN + c0 + ty) * NN + n0 + tx];
    __syncthreads();
    xT[((size_t)b * NN + n0 + ty) * CIN + c0 + tx] = s[tx][ty];
}

__global__ __launch_bounds__(128)
void dcn3d_fused_kernel(const float* __restrict__ xT,
                        const float* __restrict__ w_off,  const float* __restrict__ b_off,
                        const float* __restrict__ w_mask, const float* __restrict__ b_mask,
                        const float* __restrict__ weight, const float* __restrict__ bias,
                        float* __restrict__ out) {
    extern __shared__ float smem[];
    float* s_x   = smem;                          // 11016
    float* s_off = s_x + HALO_ROWS * SXS;         // 1904
    float* s_cw  = s_off + 112 * OFFP;            // 3456
    int*   s_ci  = (int*)(s_cw + 432 * 8);        // 3456
    float* s_a   = (float*)(s_ci + 432 * 8);      // 1088

    const int tid  = threadIdx.x;
    const int wave = tid >> 5;
    const int lane = tid & 31;
    const int lm   = lane & 15;
    const int oo   = (lane < 16) ? 0 : 2;         // K half select (ISA 16x4 f32 layout)

    const int tile = blockIdx.x;                  // 4608 tiles of 16 positions
    const int b    = tile / (NN / 16);
    const int nb   = (tile % (NN / 16)) * 16;     // flat n base
    const int z0   = nb / (HDIM * WDIM);
    const int rem  = nb % (HDIM * WDIM);
    const int y0   = rem / WDIM;
    const int x0   = rem % WDIM;                  // 0, 16, 32

    const float* xb = xT + (size_t)b * NN * CIN;

    // warm L2 for this wave's final-GEMM weight block
    __builtin_prefetch(weight + (size_t)(wave * 16 + lm) * CIN * KK, 0, 3);

    // ---- stage halo of xT (3z x 3y x 18x x 64cin) into LDS via async copies ----
    // Valid (row,chunk) elements: one GLOBAL_LOAD_ASYNC_TO_LDS_B128 each.
    // Out-of-bounds elements: ds_store_b128 zeros (disjoint addresses, so the
    // unordered DScnt/ASYNCcnt streams cannot conflict). Drain both + barrier.
    {
        const int E = HALO_ROWS * 16;             // 2592 b128 chunks
        for (int e = tid; e < E; e += 128) {
            const int hr = e >> 4;                // halo row (zz,yy,xx)
            const int ch = e & 15;                // 4-float chunk within cin
            const int zz = hr / 54, r2 = hr % 54, yy = r2 / 18, xx = r2 % 18;
            const int gz = z0 + zz - 1, gy = y0 + yy - 1, gx = x0 + xx - 1;
            float* dst = &s_x[hr * SXS + ch * 4];
            if (gz >= 0 && gz < DDIM && gy >= 0 && gy < HDIM && gx >= 0 && gx < WDIM) {
                const float* src =
                    xb + (size_t)((gz * HDIM + gy) * WDIM + gx) * CIN + ch * 4;
                async_copy_b128(src, dst);
            } else {
                const v4f zero = {0.f, 0.f, 0.f, 0.f};
                *(v4f*)dst = zero;
            }
        }
    }
    wait_async_zero();
    __syncthreads();

    // ---- offset+mask conv: [16 pos] x [108 ch] implicit GEMM, f32 WMMA ----
    for (int pass = 0; pass < 2; ++pass) {
        const int col  = pass * 64 + wave * 16 + lm;
        const int colc = (col < KCH) ? col : (KCH - 1);
        const float* wp = (colc < 81) ? (w_off + (size_t)colc * CIN * KK)
                                      : (w_mask + (size_t)(colc - 81) * CIN * KK);
        v8f acc = {0.f, 0.f, 0.f, 0.f, 0.f, 0.f, 0.f, 0.f};
        for (int t = 0; t < KK; ++t) {
            const int kd = t / 9, kh = (t / 3) % 3, kw = t % 3;
            const int rbase = ((kd * 3 + kh) * 18 + kw + lm) * SXS;
            const float* wpt = wp + t;
#pragma unroll
            for (int c = 0; c < 16; ++c) {
                v2f a = *(const v2f*)&s_x[rbase + 4 * c + oo];
                v2f bf;
                bf.x = wpt[(4 * c + oo) * KK];
                bf.y = wpt[(4 * c + oo + 1) * KK];
                acc = __builtin_amdgcn_wmma_f32_16x16x4_f32(
                    false, a, false, bf, (short)0, acc, false, false);
            }
        }
        const float bv = (colc < 81) ? b_off[colc] : b_mask[colc - 81];
        if (col < KCH) {
#pragma unroll
            for (int r = 0; r < 8; ++r) {
                const int m = r + ((lane < 16) ? 0 : 8);
                s_off[col * OFFP + m] = acc[r] + bv;
            }
        }
    }
    __syncthreads();

    // ---- softmax over the 27 mask logits per position ----
    if (tid < 16) {
        const int p = tid;
        float mx = -1e30f;
#pragma unroll
        for (int k = 0; k < KK; ++k) mx = fmaxf(mx, s_off[(81 + k) * OFFP + p]);
        float e[KK];
        float sum = 0.f;
#pragma unroll
        for (int k = 0; k < KK; ++k) {
            e[k] = __expf(s_off[(81 + k) * OFFP + p] - mx);
            sum += e[k];
        }
        const float inv = 1.f / sum;
#pragma unroll
        for (int k = 0; k < KK; ++k) s_off[(81 + k) * OFFP + p] = e[k] * inv;
    }
    __syncthreads();

    // ---- corner precompute: fold trilinear weight * valid * mask, clamp index ----
    for (int item = tid; item < KK * 16; item += 128) {
        const int k = item >> 4, p = item & 15;
        const int kd = k / 9, kh = (k / 3) % 3, kw = k % 3;
        const float oz = s_off[(k * 3 + 0) * OFFP + p];
        const float oy = s_off[(k * 3 + 1) * OFFP + p];
        const float ox = s_off[(k * 3 + 2) * OFFP + p];
        const float mk = s_off[(81 + k) * OFFP + p];
        const float pz = (float)(z0 - 1 + kd) + oz;
        const float py = (float)(y0 - 1 + kh) + oy;
        const float px = (float)(x0 + p - 1 + kw) + ox;
        const float fz = floorf(pz), fy = floorf(py), fx = floorf(px);
#pragma unroll
        for (int c = 0; c < 8; ++c) {
            const float zi = fz + (float)(c >> 2);
            const float yi = fy + (float)((c >> 1) & 1);
            const float xi = fx + (float)(c & 1);
            const float w = (1.f - fabsf(pz - zi)) * (1.f - fabsf(py - yi)) * (1.f - fabsf(px - xi));
            const bool valid = (zi >= 0.f) && (zi < (float)DDIM) && (yi >= 0.f) &&
                               (yi < (float)HDIM) && (xi >= 0.f) && (xi < (float)WDIM);
            const float zc = fminf(fmaxf(zi, 0.f), (float)(DDIM - 1));
            const float yc = fminf(fmaxf(yi, 0.f), (float)(HDIM - 1));
            const float xc = fminf(fmaxf(xi, 0.f), (float)(WDIM - 1));
            s_cw[item * 8 + c] = valid ? (w * mk) : 0.f;
            s_ci[item * 8 + c] = ((int)zc * HDIM + (int)yc) * WDIM + (int)xc;
        }
    }
    __syncthreads();

    // ---- tap loop: gather masked samples, then 16x16x64 GEMM per tap (WMMA) ----
    const int colw = wave * 16 + lm;                       // cout column
    const float* wgt = weight + (size_t)colw * CIN * KK;
    v8f acc2 = {0.f, 0.f, 0.f, 0.f, 0.f, 0.f, 0.f, 0.f};
    for (int k = 0; k < KK; ++k) {
        {   // stage A_k[16 pos][64 cin]: 8 corners, contiguous b128 reads from xT
            const int p = tid >> 3, q = tid & 7;
            const int base = (k * 16 + p) * 8;
            v4f lo = {0.f, 0.f, 0.f, 0.f}, hi = {0.f, 0.f, 0.f, 0.f};
#pragma unroll
            for (int c = 0; c < 8; ++c) {
                const float w = s_cw[base + c];
                const int idx = s_ci[base + c];
                const v4f* xp = (const v4f*)(xb + (size_t)idx * CIN + q * 8);
                lo += w * xp[0];
                hi += w * xp[1];
            }
            *(v4f*)&s_a[p * SXS + q * 8]     = lo;
            *(v4f*)&s_a[p * SXS + q * 8 + 4] = hi;
        }
        __syncthreads();
#pragma unroll
        for (int c = 0; c < 16; ++c) {
            v2f a = *(const v2f*)&s_a[lm * SXS + 4 * c + oo];
            v2f bf;
            bf.x = wgt[(4 * c + oo) * KK + k];
            bf.y = wgt[(4 * c + oo + 1) * KK + k];
            acc2 = __builtin_amdgcn_wmma_f32_16x16x4_f32(
                false, a, false, bf, (short)0, acc2, false, false);
        }
        __syncthreads();
    }

    // ---- epilogue: bias + store out[b][cout][n] ----
    const float bv2 = bias[colw];
    float* ob = out + ((size_t)b * COUT + colw) * NN + nb;
#pragma unroll
    for (int r = 0; r < 8; ++r) {
        const int m = r + ((lane < 16) ? 0 : 8);
        ob[m] = acc2[r] + bv2;
    }
}

extern "C" void kernel_launch(void* const* d_in, const int* in_sizes, int n_in,
                              void* d_out, int out_size, void* d_ws, size_t ws_size,
                              hipStream_t stream) {
    const float* x      = (const float*)d_in[0];
    const float* w_off  = (const float*)d_in[1];
    const float* b_off  = (const float*)d_in[2];
    const float* w_mask = (const float*)d_in[3];
    const float* b_mask = (const float*)d_in[4];
    const float* weight = (const float*)d_in[5];
    const float* bias   = (const float*)d_in[6];
    float* out = (float*)d_out;
    float* xT  = (float*)d_ws;   // B*N*CIN floats = 18.9 MB scratch

    transpose_x_kernel<<<BB * (NN / 16) * 4, 256, 0, stream>>>(x, xT);

    const size_t smem_bytes = (size_t)SMEM_FLOATS * sizeof(float);  // ~84 KB (<320 KB/WGP)
    dcn3d_fused_kernel<<<BB * (NN / 16), 128, smem_bytes, stream>>>(
        xT, w_off, b_off, w_mask, b_mask, weight, bias, out);
}